// CAO_91207925497926
// MI455X (gfx1250) — compile-verified
//
#include <hip/hip_runtime.h>
#include <cstdint>
#include <cmath>

typedef float v2f __attribute__((ext_vector_type(2)));
typedef float v8f __attribute__((ext_vector_type(8)));

#define M_ 16
#define N_ 32
#define P_ 1024   // N_*N_

// ---------------------------------------------------------------------------
// Kernel 0: extract permutation form of every X[a][b] (column c -> row of its 1)
// ---------------------------------------------------------------------------
__global__ __launch_bounds__(32) void CAO_perms(const float* __restrict__ X,
                                                int* __restrict__ perm)
{
    int ab = blockIdx.x;          // a*16+b
    int c  = threadIdx.x;         // column 0..31
    const float* Xab = X + (size_t)ab * (N_ * N_);
    int r1 = 0;
    #pragma unroll
    for (int r = 0; r < N_; ++r)
        if (Xab[r * N_ + c] > 0.5f) r1 = r;
    perm[ab * N_ + c] = r1;
}

// ---------------------------------------------------------------------------
// Kernel 1: per pair (i,j) build vbits (bit k of vbits[p] == vec(X[i,k]X[k,j])[p])
//           and pair_con[i,j] = overlap/512 (exact closed form of the abs-diff sum)
// ---------------------------------------------------------------------------
__global__ __launch_bounds__(256) void CAO_vbits(const int* __restrict__ perm,
                                                 uint32_t* __restrict__ vbits,
                                                 float* __restrict__ pair_con)
{
    __shared__ uint32_t vb[P_];
    __shared__ int ov;
    int pair = blockIdx.x;
    int i = pair >> 4, j = pair & 15;
    int tid = threadIdx.x;
    for (int t = tid; t < P_; t += 256) vb[t] = 0u;
    if (tid == 0) ov = 0;
    __syncthreads();
    for (int t = tid; t < M_ * N_; t += 256) {
        int k = t >> 5, c = t & 31;
        int pkj  = perm[(k * M_ + j) * N_ + c];
        int comp = perm[(i * M_ + k) * N_ + pkj];   // row of 1 in col c of X[i,k]X[k,j]
        int pij  = perm[(i * M_ + j) * N_ + c];
        if (comp == pij) atomicAdd(&ov, 1);
        atomicOr(&vb[c * N_ + comp], 1u << k);      // p = c*32 + row (column-major vec)
    }
    __syncthreads();
    for (int t = tid; t < P_; t += 256) vbits[pair * P_ + t] = vb[t];
    if (tid == 0) pair_con[pair] = (float)ov / 512.0f;
}

// ---------------------------------------------------------------------------
// Kernel 2: per pair (i,j): T = Vt * K[i,j] * V via V_WMMA_F32_16X16X4_F32,
//           g[pair][k] = T[k][k]. K streamed from HBM exactly once; V from bits.
// 8 waves; wave w owns columns [w*128, (w+1)*128) as 8 q-tiles with 8
// independent Z accumulator chains; p-loop is the only streaming loop.
// ---------------------------------------------------------------------------
__global__ __launch_bounds__(256) void CAO_quadform(const float* __restrict__ K,
                                                    const uint32_t* __restrict__ vbits,
                                                    float* __restrict__ g)
{
    __shared__ uint32_t vb[P_];          //  4 KB
    __shared__ float zscr[8][16][16];    //  8 KB per-wave Z bounce
    __shared__ float tred[8][16][16];    //  8 KB per-wave T
    int pair = blockIdx.x;
    int tid  = threadIdx.x;
    for (int t = tid; t < P_; t += 256) vb[t] = vbits[pair * P_ + t];
    __syncthreads();

    int wave = tid >> 5;       // wave32
    int lane = tid & 31;
    int hl   = lane >> 4;      // lane[4]: selects K-element pair within tile
    int l16  = lane & 15;

    const float* Kp = K + ((size_t)pair << 20);   // K[i,j], 1M floats
    int qcol = wave * 128 + l16;                  // this lane's base column

    v8f Zacc[8];
    #pragma unroll
    for (int q = 0; q < 8; ++q) Zacc[q] = (v8f){};

    // ---- streaming loop over K rows: Z[q] += Vt[:, p:p+4] * K[p:p+4, qtile] ----
    for (int p0 = 0; p0 < P_; p0 += 4) {
        int prow = p0 + hl * 2;
        // A = Vt chunk (built once per p-step): A[m=l16][kk] = bit l16 of vb[prow+kk]
        uint32_t w0 = vb[prow], w1 = vb[prow + 1];
        v2f A; A.x = (float)((w0 >> l16) & 1u);
               A.y = (float)((w1 >> l16) & 1u);
        const float* kb = Kp + (size_t)prow * P_ + qcol;
        #pragma unroll
        for (int q = 0; q < 8; ++q) {
            // B = K chunk: rows prow,prow+1 ; col qcol+16q (immediate offsets)
            v2f B; B.x = kb[q * 16]; B.y = kb[q * 16 + P_];
            Zacc[q] = __builtin_amdgcn_wmma_f32_16x16x4_f32(false, A, false, B,
                                                            (short)0, Zacc[q],
                                                            false, false);
        }
    }

    // ---- fold: T += Z[q](16x16) * V[qtile, 0:16]  (off the streaming path) ----
    v8f Tacc = {};
    #pragma unroll
    for (int qi = 0; qi < 8; ++qi) {
        // bounce Z (D-layout) -> LDS row-major (wave-private; DS in-order per wave)
        #pragma unroll
        for (int v = 0; v < 8; ++v) zscr[wave][v + 8 * hl][l16] = Zacc[qi][v];
        int qb0 = wave * 128 + qi * 16;
        #pragma unroll
        for (int qk = 0; qk < 4; ++qk) {
            v2f Az; Az.x = zscr[wave][l16][qk * 4 + hl * 2 + 0];
                    Az.y = zscr[wave][l16][qk * 4 + hl * 2 + 1];
            int q = qb0 + qk * 4 + hl * 2;
            uint32_t u0 = vb[q], u1 = vb[q + 1];
            v2f Bv; Bv.x = (float)((u0 >> l16) & 1u);
                    Bv.y = (float)((u1 >> l16) & 1u);
            Tacc = __builtin_amdgcn_wmma_f32_16x16x4_f32(false, Az, false, Bv,
                                                         (short)0, Tacc,
                                                         false, false);
        }
    }

    #pragma unroll
    for (int v = 0; v < 8; ++v) tred[wave][v + 8 * hl][l16] = Tacc[v];
    __syncthreads();
    if (tid < 16) {
        float s = 0.0f;
        #pragma unroll
        for (int w = 0; w < 8; ++w) s += tred[w][tid][tid];
        g[pair * M_ + tid] = s;   // unnormalized aff_combo; g[pair][j] == raw_aff
    }
}

// ---------------------------------------------------------------------------
// Kernel 3: norm = max over i!=j of raw_aff[i,j] = g[(i*16+j)*16 + j]
// ---------------------------------------------------------------------------
__global__ __launch_bounds__(256) void CAO_norm(const float* __restrict__ g,
                                                float* __restrict__ normp)
{
    __shared__ float red[256];
    int t = threadIdx.x;
    int j = t & 15;
    red[t] = ((t >> 4) != j) ? g[t * M_ + j] : -1e30f;
    __syncthreads();
    for (int s = 128; s > 0; s >>= 1) {
        if (t < s) red[t] = fmaxf(red[t], red[t + s]);
        __syncthreads();
    }
    if (t == 0) normp[0] = red[0];
}

// ---------------------------------------------------------------------------
// Kernel 4: per upper pair (i<j): score, first-max argmax, write X[i,j], X[j,i]^T
// ---------------------------------------------------------------------------
__global__ __launch_bounds__(32) void CAO_update(const float* __restrict__ g,
                                                 const float* __restrict__ pair_con,
                                                 const int* __restrict__ perm,
                                                 const float* __restrict__ normp,
                                                 float wa, float wc,
                                                 float* __restrict__ X)
{
    int pair = blockIdx.x;
    int i = pair >> 4, j = pair & 15;
    if (i >= j) return;                       // diag & lower handled elsewhere
    __shared__ float sc[M_];
    __shared__ int ksel;
    int lane = threadIdx.x;
    float nrm = normp[0];
    if (lane < M_) {
        float aff = g[pair * M_ + lane] / nrm;
        float con = sqrtf(pair_con[i * M_ + lane] * pair_con[lane * M_ + j]);
        sc[lane] = aff * wa + con * wc;
    }
    __syncthreads();
    if (lane == 0) {                          // jnp.argmax: first max wins
        float best = sc[0]; int bi = 0;
        for (int k = 1; k < M_; ++k)
            if (sc[k] > best) { best = sc[k]; bi = k; }
        ksel = bi;
    }
    __syncthreads();
    int ks = ksel;
    int c  = lane;                            // one thread per column
    int comp = perm[(i * M_ + ks) * N_ + perm[(ks * M_ + j) * N_ + c]];
    float* Xij = X + (size_t)pair * (N_ * N_);
    float* Xji = X + (size_t)(j * M_ + i) * (N_ * N_);
    #pragma unroll
    for (int r = 0; r < N_; ++r) {
        float v = (r == comp) ? 1.0f : 0.0f;
        Xij[r * N_ + c] = v;
        Xji[c * N_ + r] = v;
    }
}

// ---------------------------------------------------------------------------
extern "C" void kernel_launch(void* const* d_in, const int* in_sizes, int n_in,
                              void* d_out, int out_size, void* d_ws, size_t ws_size,
                              hipStream_t stream)
{
    const float* K  = (const float*)d_in[0];   // (16,16,1024,1024) f32
    const float* X0 = (const float*)d_in[1];   // (16,16,32,32) f32
    float* X = (float*)d_out;
    hipMemcpyAsync(X, X0, sizeof(float) * M_ * M_ * N_ * N_,
                   hipMemcpyDeviceToDevice, stream);

    uint8_t* w = (uint8_t*)d_ws;
    int*      perm  = (int*)w;       w += (size_t)M_ * M_ * N_ * sizeof(int);       // 32 KB
    uint32_t* vbits = (uint32_t*)w;  w += (size_t)M_ * M_ * P_ * sizeof(uint32_t);  //  1 MB
    float*    g     = (float*)w;     w += (size_t)M_ * M_ * M_ * sizeof(float);     // 16 KB
    float*    pcon  = (float*)w;     w += (size_t)M_ * M_ * sizeof(float);          //  1 KB
    float*    nrm   = (float*)w;     w += sizeof(float);

    float cst = 0.3f;
    for (int it = 0; it < 6; ++it) {
        if (it >= 2) cst = fminf(cst * 1.1f, 1.0f);
        float wa = (it < 2) ? 1.0f : (1.0f - cst);
        float wc = (it < 2) ? 0.0f : cst;
        CAO_perms   <<<M_ * M_, 32,  0, stream>>>(X, perm);
        CAO_vbits   <<<M_ * M_, 256, 0, stream>>>(perm, vbits, pcon);
        CAO_quadform<<<M_ * M_, 256, 0, stream>>>(K, vbits, g);
        CAO_norm    <<<1,       256, 0, stream>>>(g, nrm);
        CAO_update  <<<M_ * M_, 32,  0, stream>>>(g, pcon, perm, nrm, wa, wc, X);
    }
}